// TtMambaBlock_6682969113108
// MI455X (gfx1250) — compile-verified
//
#include <hip/hip_runtime.h>
#include <hip/hip_bf16.h>

#define N_TOK  128
#define DMODEL 2560
#define DINN   5120
#define SSTATE 16
#define RRANK  160
#define PCOLS  192   // R + 2*S
#define KSPLIT 16    // split-K factor for the proj GEMM
#define KCHUNK (DINN / KSPLIT)          // 320
#define NTILE_P (PCOLS / 16)            // 12
#define MTILE  (N_TOK / 16)             // 8
#define PART_STRIDE (MTILE * NTILE_P * 256)  // 24576 floats per K-chunk

typedef float v2f __attribute__((ext_vector_type(2)));
typedef float v8f __attribute__((ext_vector_type(8)));

// D = A(16x4 f32) * B(4x16 f32) + C(16x16 f32), wave32
__device__ __forceinline__ v8f wmma_k4(v2f a, v2f b, v8f c) {
  return __builtin_amdgcn_wmma_f32_16x16x4_f32(
      /*neg_a=*/false, a, /*neg_b=*/false, b,
      /*c_mod=*/(short)0, c, /*reuse_a=*/false, /*reuse_b=*/false);
}

__device__ __forceinline__ float softplusf(float v) {
  return (v > 20.f) ? v : log1pf(__expf(v));
}

// async copy 16 bytes from the global address in gaddr to the LDS address in ldsaddr  (per-lane, ASYNCcnt)
#define ASYNC_B128(ldsaddr, gaddr)                                      \
  asm volatile("global_load_async_to_lds_b128 %0, %1, off"              \
               :: "v"(ldsaddr), "v"(gaddr) : "memory")
#define WAIT_ASYNC(n) \
  asm volatile("s_wait_asynccnt " #n ::: "memory")

// ---------------------------------------------------------------------------
// Kernel 1: xs = x @ ssm_proj, res = x @ mlp_proj (shared A operand),
// then conv mix + SiLU fused on xs -> u.  One wave per 16x16 tile of DIN.
// grid = (DIN/16, N/16), block = 32.
// ---------------------------------------------------------------------------
__global__ __launch_bounds__(32) void k_gemm_xs_res(
    const float* __restrict__ x,        // (128,2560)
    const float* __restrict__ ssm_proj, // (2560,5120)
    const float* __restrict__ mlp_proj, // (2560,5120)
    const float* __restrict__ cs1, const float* __restrict__ cs2,
    const float* __restrict__ cs3,      // (128,5120) each
    const float* __restrict__ conv_w,   // (4,5120)
    const float* __restrict__ conv_b,   // (5120)
    float* __restrict__ u, float* __restrict__ res) {
  const int lane = threadIdx.x;
  const int half = lane >> 4;          // 0: K={0,1}, 1: K={2,3}
  const int l16  = lane & 15;
  const int n0 = blockIdx.x * 16;      // column tile in DIN
  const int m0 = blockIdx.y * 16;      // row tile in N

  const float* ap  = x + (m0 + l16) * DMODEL + (half << 1);
  const float* bp0 = ssm_proj + (half << 1) * DINN + (n0 + l16);
  const float* bp1 = mlp_proj + (half << 1) * DINN + (n0 + l16);

  v8f acc0 = {}; v8f acc1 = {};
  for (int k0 = 0; k0 < DMODEL; k0 += 4) {
    if (k0 + 16 < DMODEL) {
      __builtin_prefetch(bp0 + (size_t)(k0 + 16) * DINN, 0, 3);
      __builtin_prefetch(bp1 + (size_t)(k0 + 16) * DINN, 0, 3);
    }
    v2f a = *(const v2f*)(ap + k0);
    v2f b0; b0.x = bp0[(size_t)k0 * DINN]; b0.y = bp0[(size_t)(k0 + 1) * DINN];
    v2f b1; b1.x = bp1[(size_t)k0 * DINN]; b1.y = bp1[(size_t)(k0 + 1) * DINN];
    acc0 = wmma_k4(a, b0, acc0);
    acc1 = wmma_k4(a, b1, acc1);
  }

  const int col = n0 + l16;
  const float cb = conv_b[col];
  const float w0 = conv_w[col];
  const float w1 = conv_w[DINN + col];
  const float w2 = conv_w[2 * DINN + col];
  const float w3 = conv_w[3 * DINN + col];
#pragma unroll
  for (int v = 0; v < 8; ++v) {
    const int row = m0 + v + (half << 3);
    const int idx = row * DINN + col;
    const float xs = acc0[v];
    const float c  = cb + w0 * cs1[idx] + w1 * cs2[idx] + w2 * cs3[idx] + w3 * xs;
    const float uu = c / (1.f + __expf(-c));   // SiLU
    u[idx]   = uu;
    res[idx] = acc1[v];
  }
}

// ---------------------------------------------------------------------------
// Kernel 2a: split-K partials of proj = u @ x_proj_w  (128,5120)x(5120,192)
// grid = (12, 8, KSPLIT), block = 32.  Each wave: one 16x16 tile, one K-chunk.
// ---------------------------------------------------------------------------
__global__ __launch_bounds__(32) void k_proj_part(
    const float* __restrict__ u, const float* __restrict__ xpw,
    float* __restrict__ partials) {
  const int lane = threadIdx.x;
  const int half = lane >> 4;
  const int l16  = lane & 15;
  const int n0 = blockIdx.x * 16;
  const int m0 = blockIdx.y * 16;
  const int kbeg = blockIdx.z * KCHUNK;

  const float* ap = u + (m0 + l16) * DINN + kbeg + (half << 1);
  const float* bp = xpw + (size_t)(kbeg + (half << 1)) * PCOLS + (n0 + l16);

  v8f acc = {};
  for (int k0 = 0; k0 < KCHUNK; k0 += 4) {
    v2f a = *(const v2f*)(ap + k0);
    v2f b; b.x = bp[(size_t)k0 * PCOLS]; b.y = bp[(size_t)(k0 + 1) * PCOLS];
    acc = wmma_k4(a, b, acc);
  }
  const size_t base =
      (size_t)blockIdx.z * PART_STRIDE +
      ((size_t)blockIdx.y * NTILE_P + blockIdx.x) * 256;
#pragma unroll
  for (int v = 0; v < 8; ++v) partials[base + v * 32 + lane] = acc[v];
}

// Kernel 2b: reduce partials -> proj.  24576 elements, 96 blocks x 256.
__global__ __launch_bounds__(256) void k_proj_reduce(
    const float* __restrict__ partials, float* __restrict__ proj) {
  const int t = blockIdx.x * 256 + threadIdx.x;     // [0, PART_STRIDE)
  float s = 0.f;
#pragma unroll
  for (int kc = 0; kc < KSPLIT; ++kc) s += partials[(size_t)kc * PART_STRIDE + t];
  const int mn   = t >> 8;
  const int e    = t & 255;
  const int m    = mn / NTILE_P;
  const int n    = mn % NTILE_P;
  const int v    = e >> 5;
  const int lane = e & 31;
  const int row  = m * 16 + v + ((lane >> 4) << 3);
  const int col  = n * 16 + (lane & 15);
  proj[row * PCOLS + col] = s;
}

// ---------------------------------------------------------------------------
// Kernel 3: dt = softplus(proj[:, :160] @ dt_proj_w + dt_proj_b)
// grid = (5120/16, 128/16), block = 32.  K = 160.
// ---------------------------------------------------------------------------
__global__ __launch_bounds__(32) void k_gemm_dt(
    const float* __restrict__ proj, const float* __restrict__ dtw,
    const float* __restrict__ dtb, float* __restrict__ dt) {
  const int lane = threadIdx.x;
  const int half = lane >> 4;
  const int l16  = lane & 15;
  const int n0 = blockIdx.x * 16;
  const int m0 = blockIdx.y * 16;

  const float* ap = proj + (m0 + l16) * PCOLS + (half << 1);
  const float* bp = dtw + (half << 1) * DINN + (n0 + l16);

  v8f acc = {};
  for (int k0 = 0; k0 < RRANK; k0 += 4) {
    v2f a = *(const v2f*)(ap + k0);
    v2f b; b.x = bp[(size_t)k0 * DINN]; b.y = bp[(size_t)(k0 + 1) * DINN];
    acc = wmma_k4(a, b, acc);
  }
  const int col = n0 + l16;
  const float bias = dtb[col];
#pragma unroll
  for (int v = 0; v < 8; ++v) {
    const int row = m0 + v + (half << 3);
    dt[row * DINN + col] = softplusf(acc[v] + bias);
  }
}

// ---------------------------------------------------------------------------
// Kernel 4: SSM step + gating.  z[n,d] = (sum_s e^{dt A} state C + dt*u*(B.C)
//            + D*u) * res.   grid = (DIN/256, N), block = 256.
// ---------------------------------------------------------------------------
__global__ __launch_bounds__(256) void k_ssm(
    const float* __restrict__ proj, const float* __restrict__ dt,
    const float* __restrict__ u, const float* __restrict__ res,
    const float* __restrict__ ssm_state, const float* __restrict__ A_log,
    const float* __restrict__ Dp, float* __restrict__ z) {
  __shared__ float sC[SSTATE];
  __shared__ float sBC;
  const int n = blockIdx.y;
  const int d = blockIdx.x * 256 + threadIdx.x;

  if (threadIdx.x < SSTATE)
    sC[threadIdx.x] = proj[n * PCOLS + RRANK + SSTATE + threadIdx.x];
  if (threadIdx.x == 0) {
    float bc = 0.f;
#pragma unroll
    for (int s = 0; s < SSTATE; ++s)
      bc += proj[n * PCOLS + RRANK + s] * proj[n * PCOLS + RRANK + SSTATE + s];
    sBC = bc;
  }
  __syncthreads();

  const float dtv = dt[n * DINN + d];
  const float uv  = u[n * DINN + d];
  const float* st = ssm_state + ((size_t)n * DINN + d) * SSTATE;
  const float* al = A_log + (size_t)d * SSTATE;

  float y = 0.f;
#pragma unroll
  for (int s = 0; s < SSTATE; ++s) {
    const float a = -__expf(al[s]);
    y += __expf(dtv * a) * st[s] * sC[s];
  }
  y += dtv * uv * sBC + Dp[d] * uv;
  z[n * DINN + d] = y * res[n * DINN + d];
}

// ---------------------------------------------------------------------------
// Kernel 5: out = z @ down_proj  (128,5120)x(5120,2560).
// Cooperative: block = 256 (8 waves, one M-tile each) per N-tile.
// Wave 0 stages the shared 8x16 B-tile into LDS with async-to-LDS (double
// buffered, ASYNCcnt + workgroup barrier); all waves feed WMMA from LDS.
// grid = (2560/16), block = 256.
// ---------------------------------------------------------------------------
__global__ __launch_bounds__(256) void k_gemm_out_coop(
    const float* __restrict__ zin, const float* __restrict__ dwn,
    float* __restrict__ out) {
  __shared__ float bbuf[2][128];       // 8 K-rows x 16 cols per stage
  const int tid  = threadIdx.x;
  const int wave = tid >> 5;           // 0..7 -> M-tile
  const int lane = tid & 31;
  const int half = lane >> 4;
  const int l16  = lane & 15;
  const int n0 = blockIdx.x * 16;
  const int m0 = wave * 16;

  const float* ap = zin + (m0 + l16) * DINN + (half << 1);

  // wave-0 async loader mapping: lane -> (K-row, 4-dword column group)
  const int lrow = lane >> 2;          // 0..7
  const int lcol = (lane & 3) << 2;    // 0,4,8,12
  const float* gB = dwn + (size_t)lrow * DMODEL + n0 + lcol;
  const unsigned lds0 = (unsigned)(size_t)&bbuf[0][lrow * 16 + lcol];
  const unsigned lds1 = (unsigned)(size_t)&bbuf[1][lrow * 16 + lcol];

  if (wave == 0) {
    ASYNC_B128(lds0, (unsigned long long)(size_t)gB);
    WAIT_ASYNC(0x0);
  }
  __syncthreads();

  v8f acc = {};
  for (int k0 = 0; k0 < DINN; k0 += 8) {
    const int cur = (k0 >> 3) & 1;
    if (wave == 0 && k0 + 8 < DINN) {
      ASYNC_B128(cur ? lds0 : lds1,
                 (unsigned long long)(size_t)(gB + (size_t)(k0 + 8) * DMODEL));
    }
    const float* bb = &bbuf[cur][0];
#pragma unroll
    for (int kk = 0; kk < 8; kk += 4) {
      v2f a = *(const v2f*)(ap + k0 + kk);
      v2f b;
      b.x = bb[(kk + (half << 1)) * 16 + l16];
      b.y = bb[(kk + (half << 1) + 1) * 16 + l16];
      acc = wmma_k4(a, b, acc);
    }
    if (wave == 0 && k0 + 8 < DINN) WAIT_ASYNC(0x0);
    __syncthreads();
  }

  const int col = n0 + l16;
#pragma unroll
  for (int v = 0; v < 8; ++v) {
    const int row = m0 + v + (half << 3);
    out[row * DMODEL + col] = acc[v];
  }
}

// ---------------------------------------------------------------------------
extern "C" void kernel_launch(void* const* d_in, const int* in_sizes, int n_in,
                              void* d_out, int out_size, void* d_ws, size_t ws_size,
                              hipStream_t stream) {
  const float* x         = (const float*)d_in[0];
  // d_in[1] = conv_state0 : unused by the reference math
  const float* cs1       = (const float*)d_in[2];
  const float* cs2       = (const float*)d_in[3];
  const float* cs3       = (const float*)d_in[4];
  const float* ssm_state = (const float*)d_in[5];
  const float* ssm_proj  = (const float*)d_in[6];
  const float* mlp_proj  = (const float*)d_in[7];
  const float* down_proj = (const float*)d_in[8];
  const float* conv_w    = (const float*)d_in[9];
  const float* conv_b    = (const float*)d_in[10];
  const float* x_proj_w  = (const float*)d_in[11];
  const float* dt_proj_w = (const float*)d_in[12];
  const float* dt_proj_b = (const float*)d_in[13];
  const float* A_log     = (const float*)d_in[14];
  const float* Dp        = (const float*)d_in[15];
  float* out = (float*)d_out;

  // workspace layout (floats): u, res, proj, dt, z, partials
  float* ws    = (float*)d_ws;
  float* u     = ws;
  float* res   = u    + (size_t)N_TOK * DINN;
  float* proj  = res  + (size_t)N_TOK * DINN;
  float* dtb   = proj + (size_t)N_TOK * PCOLS;
  float* z     = dtb  + (size_t)N_TOK * DINN;
  float* parts = z    + (size_t)N_TOK * DINN;

  k_gemm_xs_res<<<dim3(DINN / 16, N_TOK / 16), 32, 0, stream>>>(
      x, ssm_proj, mlp_proj, cs1, cs2, cs3, conv_w, conv_b, u, res);

  k_proj_part<<<dim3(NTILE_P, MTILE, KSPLIT), 32, 0, stream>>>(u, x_proj_w, parts);
  k_proj_reduce<<<dim3(PART_STRIDE / 256), 256, 0, stream>>>(parts, proj);

  k_gemm_dt<<<dim3(DINN / 16, N_TOK / 16), 32, 0, stream>>>(proj, dt_proj_w, dt_proj_b, dtb);

  k_ssm<<<dim3(DINN / 256, N_TOK), 256, 0, stream>>>(
      proj, dtb, u, res, ssm_state, A_log, Dp, z);

  k_gemm_out_coop<<<dim3(DMODEL / 16), 256, 0, stream>>>(z, down_proj, out);
}